// QuantumCircuit_66726611910954
// MI455X (gfx1250) — compile-verified
//
#include <hip/hip_runtime.h>
#include <hip/hip_bf16.h>

// ---------------------------------------------------------------------------
// 20-qubit state-vector simulator for MI455X (gfx1250, wave32).
// Qubit q <-> flat-index bit (19 - q).  Any 4-bit index field [S, S+4) turns
// the state into a 16 x 65536 matrix (k-stride 2^S), so 4-qubit gate groups
// apply as 16x16 complex GEMMs on V_WMMA_F32_16X16X4_F32.
// Per block (14 full-state passes, 13 on the matrix pipe):
//   G0 = cu3triple0 @ kron(u3[0..3])  (S=16)   [triple0 commutes w/ u3 4..19]
//   kron(u3[4a..4a+3])                (S=16-4a, a=1..4)
//   cross(4a+3,4a+4)                  (S=14-4a, a=0..3)  -- 16x16 WMMA
//   cu3triple a                       (S=16-4a, a=1..4)
//   cu3(19,0)                         scalar pair kernel (bits 0,19)
// Pure L2-bandwidth problem: passes = time; 14 passes vs naive 40.
// ---------------------------------------------------------------------------

typedef float v2f __attribute__((ext_vector_type(2)));
typedef float v8f __attribute__((ext_vector_type(8)));

#define NQ 20
#define NSTATES (1u << NQ)      // 1048576
#define NCPAIRS (1u << (NQ-2))  // 262144

__device__ __forceinline__ float2 cmul(float2 a, float2 b) {
    return make_float2(a.x*b.x - a.y*b.y, a.x*b.y + a.y*b.x);
}
__device__ __forceinline__ float2 cadd(float2 a, float2 b) {
    return make_float2(a.x + b.x, a.y + b.y);
}

// Element (r,s) of the 16x16 matrix of a controlled-U acting inside a 4-bit
// field: control bit bc, target bit bt, u = 2x2 target unitary (row-major).
__device__ __forceinline__ float2 cu3_elem(const float2* u, int bc, int bt,
                                           int r, int s) {
    unsigned tm = (~(1u << bt)) & 15u;
    if ((r & tm) != (s & tm)) return make_float2(0.f, 0.f);
    if (((r >> bc) & 1) == 0)
        return (r == s) ? make_float2(1.f, 0.f) : make_float2(0.f, 0.f);
    return u[((r >> bt) & 1) * 2 + ((s >> bt) & 1)];
}

// ---------------------------------------------------------------------------
// Prep: 120 2x2 gate matrices from params, then per (block k, nibble a):
//   wu3 [k][a] = kron(U_{4a},..,U_{4a+3})                       (16x16)
//   wcu3[k][a] = M(4a+2,4a+3) @ M(4a+1,4a+2) @ M(4a,4a+1)       (16x16)
//   wu3 [k][0] <- wcu3[k][0] @ wu3[k][0]   (merged G0)
//   wcross[k][a] (a<4): cu3(4a+3,4a+4) inside field S=14-4a -> (bc,bt)=(2,1)
// Within a nibble field, qubit (4a+j) maps to field bit (3-j), so the cu3
// triple always has (bc,bt) = (3,2),(2,1),(1,0).  One block of 256 threads.
// ---------------------------------------------------------------------------
__global__ void prep_kernel(const float* __restrict__ u3p,
                            const float* __restrict__ cu3p,
                            float2* __restrict__ u3m,    // 60 * 4
                            float2* __restrict__ cu3u,   // 60 * 4
                            float2* __restrict__ wu3,    // 15 * 256
                            float2* __restrict__ wcu3,   // 15 * 256
                            float2* __restrict__ wcross) // 12 * 256
{
    const int t = threadIdx.x;
    if (t < 120) {
        const float* p   = (t < 60) ? (u3p + 3 * t) : (cu3p + 3 * (t - 60));
        float2*      dst = (t < 60) ? (u3m + 4 * t) : (cu3u + 4 * (t - 60));
        float th = p[0], ph = p[1], la = p[2];
        float ct = cosf(0.5f * th), st = sinf(0.5f * th);
        dst[0] = make_float2(ct, 0.f);
        dst[1] = make_float2(-cosf(la) * st, -sinf(la) * st);
        dst[2] = make_float2( cosf(ph) * st,  sinf(ph) * st);
        dst[3] = make_float2(cosf(ph + la) * ct, sinf(ph + la) * ct);
    }
    __syncthreads();

    __shared__ float2 Tsh[256];
    __shared__ float2 Ush[256];
    const int r = t >> 4, c = t & 15;
    for (int k = 0; k < 3; ++k) {
        for (int a = 0; a < 5; ++a) {
            const int qb  = 4 * a;
            const int idx = (k * 5 + a) * 256 + t;
            // kron of the 4 u3 gates of this nibble
            float2 w = make_float2(1.f, 0.f);
            #pragma unroll
            for (int j = 0; j < 4; ++j) {
                int pos = 3 - j;
                int rb = (r >> pos) & 1, cb = (c >> pos) & 1;
                w = cmul(w, u3m[(k * 20 + qb + j) * 4 + rb * 2 + cb]);
            }
            wu3[idx] = w;

            // nibble-crossing cu3 (qb+3, qb+4) inside field S = 14-4a
            if (a < 4)
                wcross[(k * 4 + a) * 256 + t] =
                    cu3_elem(cu3u + (k * 20 + qb + 3) * 4, 2, 1, r, c);

            // ordered product of the in-nibble cu3 triple
            const float2* u1 = cu3u + (k * 20 + qb + 0) * 4; // bc=3, bt=2
            const float2* u2 = cu3u + (k * 20 + qb + 1) * 4; // bc=2, bt=1
            const float2* u3 = cu3u + (k * 20 + qb + 2) * 4; // bc=1, bt=0
            float2 acc = make_float2(0.f, 0.f);
            for (int s = 0; s < 16; ++s)
                acc = cadd(acc, cmul(cu3_elem(u2, 2, 1, r, s),
                                     cu3_elem(u1, 3, 2, s, c)));
            Tsh[t] = acc;
            __syncthreads();
            acc = make_float2(0.f, 0.f);
            for (int s = 0; s < 16; ++s)
                acc = cadd(acc, cmul(cu3_elem(u3, 1, 0, r, s),
                                     Tsh[s * 16 + c]));
            wcu3[idx] = acc;
            __syncthreads();

            if (a == 0) {   // merged first gate: G0 = triple0 @ u3nibble0
                Ush[t] = w;
                Tsh[t] = acc;
                __syncthreads();
                float2 g = make_float2(0.f, 0.f);
                for (int s = 0; s < 16; ++s)
                    g = cadd(g, cmul(Tsh[r * 16 + s], Ush[s * 16 + c]));
                wu3[idx] = g;
                __syncthreads();
            }
        }
    }
}

__global__ void init_psi_kernel(float2* __restrict__ psi) {
    unsigned i = blockIdx.x * blockDim.x + threadIdx.x;
    psi[i] = make_float2((i == 0) ? 1.f : 0.f, 0.f);
}

// ---------------------------------------------------------------------------
// Controlled-U on (control bit bc, target bit bt) -- only used for cu3(19,0).
// ---------------------------------------------------------------------------
__global__ void apply2_kernel(float2* __restrict__ psi,
                              const float2* __restrict__ g, int bc, int bt) {
    unsigned t = blockIdx.x * blockDim.x + threadIdx.x; // < 2^18
    int bl = (bc < bt) ? bc : bt;
    int bh = (bc < bt) ? bt : bc;
    unsigned lm = (1u << bl) - 1u;
    unsigned x  = ((t & ~lm) << 1) | (t & lm);          // insert 0 at bl
    unsigned hm = (1u << bh) - 1u;
    unsigned idx = ((x & ~hm) << 1) | (x & hm);         // insert 0 at bh
    unsigned i0 = idx | (1u << bc);                     // control = 1
    unsigned i1 = i0 | (1u << bt);
    float2 m00 = g[0], m01 = g[1], m10 = g[2], m11 = g[3];
    float2 a0 = psi[i0], a1 = psi[i1];
    psi[i0] = cadd(cmul(m00, a0), cmul(m01, a1));
    psi[i1] = cadd(cmul(m10, a0), cmul(m11, a1));
}

// ---------------------------------------------------------------------------
// Fused 16x16 complex gate on index field [SHIFT, SHIFT+4) via WMMA f32
// 16x16x4.  Column n of the 16 x 65536 view: h = n >> SHIFT,
// l = n & (2^SHIFT - 1); element(k) = (h << (SHIFT+4)) + (k << SHIFT) + l.
// Each wave owns 16 consecutive columns (same h, consecutive l -> coalesced).
//   Cr = Wr*Pr - Wi*Pi ; Ci = Wr*Pi + Wi*Pr  (16 chained K=4 WMMAs)
// In-place safe: all of a wave's loads precede its stores; column sets are
// wave-exclusive.  Grids fully populated -> EXEC all ones as WMMA requires.
// ---------------------------------------------------------------------------
template <int SHIFT>
__global__ void apply16_wmma_kernel(float2* __restrict__ psi,
                                    const float2* __restrict__ W) {
    __shared__ float2 Wsh[256];
    const int tid = threadIdx.x;
    Wsh[tid] = W[tid];
    __syncthreads();

    const int lane = tid & 31;
    const int wave = tid >> 5;
    const int nsub = lane & 15;
    const int hi   = lane >> 4;               // 0: K=k0,k0+1  1: K=k0+2,k0+3
    const unsigned colBase = (blockIdx.x * 8 + wave) * 16;

    // A fragments (W, 16x4 f32 per K-slice); f32 WMMA has no A-negate
    // (NEG is CNeg-only), so -Wi is negated in VALU here.
    v2f Ar[4], Ai[4], AiN[4];
    #pragma unroll
    for (int s = 0; s < 4; ++s) {
        int kk = 4 * s + 2 * hi;
        float2 w0 = Wsh[nsub * 16 + kk];
        float2 w1 = Wsh[nsub * 16 + kk + 1];
        Ar[s]  = (v2f){ w0.x,  w1.x};
        Ai[s]  = (v2f){ w0.y,  w1.y};
        AiN[s] = (v2f){-w0.y, -w1.y};
    }

    // B fragments from this lane's column
    const unsigned col = colBase + nsub;
    const unsigned h   = col >> SHIFT;
    const unsigned l   = col & ((1u << SHIFT) - 1u);
    float2* base = psi + ((size_t)h << (SHIFT + 4)) + l;
    v2f Br[4], Bi[4];
    #pragma unroll
    for (int s = 0; s < 4; ++s) {
        int kk = 4 * s + 2 * hi;
        float2 b0 = base[(unsigned)kk << SHIFT];
        float2 b1 = base[(unsigned)(kk + 1) << SHIFT];
        Br[s] = (v2f){b0.x, b1.x};
        Bi[s] = (v2f){b0.y, b1.y};
    }

    v8f Cr = {}; v8f Ci = {};
    #pragma unroll
    for (int s = 0; s < 4; ++s) {
        Cr = __builtin_amdgcn_wmma_f32_16x16x4_f32(false, Ar[s],  false, Br[s],
                                                   (short)0, Cr, false, false);
        Cr = __builtin_amdgcn_wmma_f32_16x16x4_f32(false, AiN[s], false, Bi[s],
                                                   (short)0, Cr, false, false);
        Ci = __builtin_amdgcn_wmma_f32_16x16x4_f32(false, Ar[s],  false, Bi[s],
                                                   (short)0, Ci, false, false);
        Ci = __builtin_amdgcn_wmma_f32_16x16x4_f32(false, Ai[s],  false, Br[s],
                                                   (short)0, Ci, false, false);
    }

    // D layout: lane<16 holds M=0..7 of its column, lane>=16 holds M=8..15
    const int mBase = hi * 8;
    #pragma unroll
    for (int v = 0; v < 8; ++v)
        base[(unsigned)(mBase + v) << SHIFT] = make_float2(Cr[v], Ci[v]);
}

// ---------------------------------------------------------------------------
// probs -> x = (0.8*tanh(0.1*2^19*p))^0.3, deterministic two-stage mean
// ---------------------------------------------------------------------------
__global__ void transform_kernel(const float2* __restrict__ psi,
                                 float* __restrict__ out,
                                 float* __restrict__ partials) {
    __shared__ float sdata[256];
    unsigned i = blockIdx.x * 256 + threadIdx.x;
    float2 a = psi[i];
    float p = a.x * a.x + a.y * a.y;
    float x = powf(0.8f * tanhf(0.1f * 524288.0f * p), 0.3f);
    out[i] = x;
    sdata[threadIdx.x] = x;
    __syncthreads();
    for (int s = 128; s > 0; s >>= 1) {
        if (threadIdx.x < s) sdata[threadIdx.x] += sdata[threadIdx.x + s];
        __syncthreads();
    }
    if (threadIdx.x == 0) partials[blockIdx.x] = sdata[0];
}

__global__ void reduce_mean_kernel(const float* __restrict__ partials,
                                   float* __restrict__ meanBuf) {
    __shared__ float sdata[256];
    float s = 0.f;
    for (int j = threadIdx.x; j < 4096; j += 256) s += partials[j];
    sdata[threadIdx.x] = s;
    __syncthreads();
    for (int k = 128; k > 0; k >>= 1) {
        if (threadIdx.x < k) sdata[threadIdx.x] += sdata[threadIdx.x + k];
        __syncthreads();
    }
    if (threadIdx.x == 0) meanBuf[0] = sdata[0] * (1.f / 1048576.f);
}

__global__ void subtract_mean_kernel(float* __restrict__ out,
                                     const float* __restrict__ meanBuf) {
    unsigned i = blockIdx.x * 256 + threadIdx.x;
    out[i] -= meanBuf[0];
}

// ---------------------------------------------------------------------------
static inline void launch_apply16(int shift, float2* psi, const float2* W,
                                  hipStream_t stream) {
    switch (shift) {
        case 0:  apply16_wmma_kernel<0><<<512, 256, 0, stream>>>(psi, W);  break;
        case 2:  apply16_wmma_kernel<2><<<512, 256, 0, stream>>>(psi, W);  break;
        case 4:  apply16_wmma_kernel<4><<<512, 256, 0, stream>>>(psi, W);  break;
        case 6:  apply16_wmma_kernel<6><<<512, 256, 0, stream>>>(psi, W);  break;
        case 8:  apply16_wmma_kernel<8><<<512, 256, 0, stream>>>(psi, W);  break;
        case 10: apply16_wmma_kernel<10><<<512, 256, 0, stream>>>(psi, W); break;
        case 12: apply16_wmma_kernel<12><<<512, 256, 0, stream>>>(psi, W); break;
        case 14: apply16_wmma_kernel<14><<<512, 256, 0, stream>>>(psi, W); break;
        default: apply16_wmma_kernel<16><<<512, 256, 0, stream>>>(psi, W); break;
    }
}

extern "C" void kernel_launch(void* const* d_in, const int* in_sizes, int n_in,
                              void* d_out, int out_size, void* d_ws,
                              size_t ws_size, hipStream_t stream) {
    const float* u3p  = (const float*)d_in[0];   // (3,20,3)
    const float* cu3p = (const float*)d_in[1];   // (3,20,3)
    float* out = (float*)d_out;                  // 2^20 floats
    float* ws  = (float*)d_ws;

    float2* psi    = (float2*)ws;                        // 2^20 complex (8 MB)
    float2* u3m    = (float2*)(ws + (1u << 21));         // 60 * 4
    float2* cu3u   = u3m + 240;                          // 60 * 4
    float2* wu3    = cu3u + 240;                         // 15 * 256
    float2* wcu3   = wu3 + 3840;                         // 15 * 256
    float2* wcross = wcu3 + 3840;                        // 12 * 256
    float*  partials = (float*)(wcross + 3072);          // 4096
    float*  meanBuf  = partials + 4096;                  // 1

    prep_kernel<<<1, 256, 0, stream>>>(u3p, cu3p, u3m, cu3u, wu3, wcu3, wcross);
    init_psi_kernel<<<NSTATES / 256, 256, 0, stream>>>(psi);

    for (int k = 0; k < 3; ++k) {
        // merged G0 = cu3triple0 @ kron(u3[0..3]) on field S=16 (legal: the
        // triple commutes with u3 gates on qubits 4..19)
        launch_apply16(16, psi, wu3 + (k * 5 + 0) * 256, stream);
        // remaining u3 nibbles (u3 gates all commute)
        for (int a = 1; a < 5; ++a)
            launch_apply16(16 - 4 * a, psi, wu3 + (k * 5 + a) * 256, stream);
        // cu3 ring remainder in reference order:
        //   cross(4a+3,4a+4) on S=14-4a, then triple a+1 on S=16-4(a+1)
        for (int a = 0; a < 4; ++a) {
            launch_apply16(14 - 4 * a, psi, wcross + (k * 4 + a) * 256, stream);
            launch_apply16(16 - 4 * (a + 1), psi,
                           wcu3 + (k * 5 + a + 1) * 256, stream);
        }
        // cu3 (19, 0): bits (0, 19) are non-contiguous -> scalar pair pass
        apply2_kernel<<<NCPAIRS / 256, 256, 0, stream>>>(
            psi, cu3u + (k * 20 + 19) * 4, 0, 19);
    }

    transform_kernel<<<NSTATES / 256, 256, 0, stream>>>(psi, out, partials);
    reduce_mean_kernel<<<1, 256, 0, stream>>>(partials, meanBuf);
    subtract_mean_kernel<<<NSTATES / 256, 256, 0, stream>>>(out, meanBuf);
}